// Model_32409823216261
// MI455X (gfx1250) — compile-verified
//
#include <hip/hip_runtime.h>
#include <hip/hip_bf16.h>
#include <math.h>

#define TPB 256
#define Mn 12000
#define Dn 8000
#define FGn 256
#define EMn 384000
#define EDn 256000

typedef __attribute__((ext_vector_type(2))) float v2f;
typedef __attribute__((ext_vector_type(8))) float v8f;

__device__ __forceinline__ float lrelu_(float v, float s) { return v >= 0.f ? v : s * v; }

__device__ __forceinline__ void atomicMaxF(float* addr, float v) {
    if (v >= 0.f) atomicMax((int*)addr, __float_as_int(v));
    else          atomicMin((unsigned int*)addr, __float_as_uint(v));
}

// ---------------------------------------------------------------- elementwise
__global__ void k_fill(float* __restrict__ p, float v, int n) {
    int i = blockIdx.x * blockDim.x + threadIdx.x;
    if (i < n) p[i] = v;
}

__global__ void k_deg(const int* __restrict__ dst, const float* __restrict__ w,
                      float* __restrict__ deg, int E) {
    int e = blockIdx.x * blockDim.x + threadIdx.x;
    if (e < E) atomicAdd(&deg[dst[e]], w[e]);
}

__global__ void k_dinv(float* __restrict__ p, int n) {
    int i = blockIdx.x * blockDim.x + threadIdx.x;
    if (i < n) { float d = p[i]; p[i] = d > 0.f ? rsqrtf(d) : 0.f; }
}

__global__ void k_bias_act(float* __restrict__ x, const float* __restrict__ b,
                           int total, int F) {
    int i = blockIdx.x * blockDim.x + threadIdx.x;
    if (i >= total) return;
    float v = x[i] + b[i % F];
    x[i] = lrelu_(v, 0.01f);
}

// A = b0*X1 + b1*A + (1-b0-b1)*X3   (A holds X2 on entry)
__global__ void k_combine(float* __restrict__ A, const float* __restrict__ X1,
                          const float* __restrict__ X3, const float* __restrict__ beta,
                          int total) {
    int i = blockIdx.x * blockDim.x + threadIdx.x;
    if (i >= total) return;
    float b0 = beta[0], b1 = beta[1];
    A[i] = b0 * X1[i] + b1 * A[i] + (1.f - b0 - b1) * X3[i];
}

// ---------------------------------------------------------------- WMMA GEMMs
// C[n,F] = A[n,K] @ W[K,F] (+bias, +lrelu0.01).  One 16x16 tile per wave32.
__global__ void __launch_bounds__(TPB)
gemm_wmma_nn(const float* __restrict__ A, const float* __restrict__ W,
             const float* __restrict__ bias, float* __restrict__ Cout,
             int n, int K, int F, int act) {
    int wave = threadIdx.x >> 5;
    int lane = threadIdx.x & 31;
    int tile = blockIdx.x * (blockDim.x >> 5) + wave;
    int tcols = F >> 4;
    int total = (n >> 4) * tcols;
    if (tile >= total) return;                   // wave-uniform: EXEC stays full
    int row0 = (tile / tcols) << 4;
    int col0 = (tile % tcols) << 4;
    int lo = lane & 15, hi = lane >> 4;

    const float* arow = A + (size_t)(row0 + lo) * K + 2 * hi;   // A frag: lane row, K pair
    const float* bcol = W + (size_t)(2 * hi) * F + col0 + lo;   // B frag: lane col, K pair
    v8f c = {};
#pragma unroll 4
    for (int k = 0; k < K; k += 4) {
        v2f a = *(const v2f*)(arow + k);
        v2f b;
        b.x = bcol[(size_t)k * F];
        b.y = bcol[(size_t)k * F + F];
        c = __builtin_amdgcn_wmma_f32_16x16x4_f32(false, a, false, b, (short)0, c, false, false);
    }
#pragma unroll
    for (int v = 0; v < 8; ++v) {
        int r = row0 + v + 8 * hi;
        int cc = col0 + lo;
        float x = c[v];
        if (bias) x += bias[cc];
        if (act) x = lrelu_(x, 0.01f);
        Cout[(size_t)r * F + cc] = x;
    }
}

// Out[Mo,No] = X[Mo,K] @ Y[No,K]^T
__global__ void __launch_bounds__(TPB)
gemm_wmma_nt(const float* __restrict__ X, const float* __restrict__ Y,
             float* __restrict__ Out, int Mo, int No, int K) {
    int wave = threadIdx.x >> 5;
    int lane = threadIdx.x & 31;
    int tile = blockIdx.x * (blockDim.x >> 5) + wave;
    int tcols = No >> 4;
    int total = (Mo >> 4) * tcols;
    if (tile >= total) return;
    int row0 = (tile / tcols) << 4;
    int col0 = (tile % tcols) << 4;
    int lo = lane & 15, hi = lane >> 4;

    const float* arow = X + (size_t)(row0 + lo) * K + 2 * hi;
    const float* brow = Y + (size_t)(col0 + lo) * K + 2 * hi;   // B[k][n] = Y[n][k]
    v8f c = {};
#pragma unroll 4
    for (int k = 0; k < K; k += 4) {
        v2f a = *(const v2f*)(arow + k);
        v2f b = *(const v2f*)(brow + k);
        c = __builtin_amdgcn_wmma_f32_16x16x4_f32(false, a, false, b, (short)0, c, false, false);
    }
#pragma unroll
    for (int v = 0; v < 8; ++v) {
        int r = row0 + v + 8 * hi;
        Out[(size_t)r * No + col0 + lo] = c[v];
    }
}

// ---------------------------------------------------------------- GCN scatter
// out[i,:] = dinv[i]^2 * h[i,:]   (self-loop contribution initializes out)
__global__ void k_gcn_self(const float* __restrict__ h, const float* __restrict__ dinv,
                           float* __restrict__ out, int n) {
    int i = blockIdx.x * blockDim.x + threadIdx.x;
    if (i >= n * 64) return;
    int node = i >> 6;
    int f4 = (i & 63) << 2;
    float s = dinv[node]; s = s * s;
    float4 hv = *(const float4*)(h + (size_t)node * FGn + f4);
    float4 o = make_float4(s * hv.x, s * hv.y, s * hv.z, s * hv.w);
    *(float4*)(out + (size_t)node * FGn + f4) = o;
}

// out[d,:] += dinv[s]*w*dinv[d] * h[s,:]
__global__ void k_gcn_scatter(const int* __restrict__ src, const int* __restrict__ dst,
                              const float* __restrict__ w, const float* __restrict__ dinv,
                              const float* __restrict__ h, float* __restrict__ out, int E) {
    int i = blockIdx.x * blockDim.x + threadIdx.x;
    if (i >= E * 64) return;
    int e = i >> 6;
    int f4 = (i & 63) << 2;
    int s = src[e], d = dst[e];
    float nrm = dinv[s] * w[e] * dinv[d];
    float4 hv = *(const float4*)(h + (size_t)s * FGn + f4);
    float* o = out + (size_t)d * FGn + f4;
    atomicAdd(o + 0, nrm * hv.x);
    atomicAdd(o + 1, nrm * hv.y);
    atomicAdd(o + 2, nrm * hv.z);
    atomicAdd(o + 3, nrm * hv.w);
}

// ---------------------------------------------------------------- GAT kernels
__global__ void k_att(const float* __restrict__ hb, const float* __restrict__ aS,
                      const float* __restrict__ aD, float* __restrict__ as_,
                      float* __restrict__ ad_, int n) {
    int i = blockIdx.x * blockDim.x + threadIdx.x;
    if (i >= n * 4) return;
    int node = i >> 2, h = i & 3;
    const float* hp = hb + (size_t)node * FGn + h * 64;
    const float* sp = aS + h * 64;
    const float* dp = aD + h * 64;
    float s0 = 0.f, s1 = 0.f;
#pragma unroll 8
    for (int k = 0; k < 64; ++k) { float v = hp[k]; s0 += v * sp[k]; s1 += v * dp[k]; }
    as_[i] = s0; ad_[i] = s1;
}

// emax initialized with the self-loop logit (always present)
__global__ void k_emax_init(const float* __restrict__ as_, const float* __restrict__ ad_,
                            float* __restrict__ emax, int n) {
    int i = blockIdx.x * blockDim.x + threadIdx.x;
    if (i < n * 4) emax[i] = lrelu_(as_[i] + ad_[i], 0.2f);
}

__global__ void k_edge_max(const int* __restrict__ src, const int* __restrict__ dst,
                           const float* __restrict__ as_, const float* __restrict__ ad_,
                           float* __restrict__ emax, int E) {
    int e = blockIdx.x * blockDim.x + threadIdx.x;
    if (e >= E) return;
    int s = src[e], d = dst[e];
#pragma unroll
    for (int h = 0; h < 4; ++h) {
        float v = lrelu_(as_[s * 4 + h] + ad_[d * 4 + h], 0.2f);
        atomicMaxF(&emax[d * 4 + h], v);
    }
}

// denom initialized with self-loop term; exs caches it
__global__ void k_self_exp(const float* __restrict__ as_, const float* __restrict__ ad_,
                           const float* __restrict__ emax, float* __restrict__ exs,
                           float* __restrict__ den, int n) {
    int i = blockIdx.x * blockDim.x + threadIdx.x;
    if (i >= n * 4) return;
    float e = lrelu_(as_[i] + ad_[i], 0.2f);
    float ex = expf(e - emax[i]);
    exs[i] = ex;
    den[i] = ex;
}

__global__ void k_edge_exp(const int* __restrict__ src, const int* __restrict__ dst,
                           const float* __restrict__ as_, const float* __restrict__ ad_,
                           const float* __restrict__ emax, float* __restrict__ exE,
                           float* __restrict__ den, int E) {
    int e = blockIdx.x * blockDim.x + threadIdx.x;
    if (e >= E) return;
    int s = src[e], d = dst[e];
#pragma unroll
    for (int h = 0; h < 4; ++h) {
        float v = lrelu_(as_[s * 4 + h] + ad_[d * 4 + h], 0.2f);
        float ex = expf(v - emax[d * 4 + h]);
        exE[e * 4 + h] = ex;
        atomicAdd(&den[d * 4 + h], ex);
    }
}

// out[i,:] = (exs/den) * hb[i,:]   (self-loop contribution initializes out)
__global__ void k_self_agg(const float* __restrict__ hb, const float* __restrict__ exs,
                           const float* __restrict__ den, float* __restrict__ out, int n) {
    int i = blockIdx.x * blockDim.x + threadIdx.x;
    if (i >= n * 64) return;
    int node = i >> 6;
    int f4 = (i & 63) << 2;
    int h = f4 >> 6;
    float coef = exs[node * 4 + h] / den[node * 4 + h];
    float4 hv = *(const float4*)(hb + (size_t)node * FGn + f4);
    float4 o = make_float4(coef * hv.x, coef * hv.y, coef * hv.z, coef * hv.w);
    *(float4*)(out + (size_t)node * FGn + f4) = o;
}

__global__ void k_edge_agg(const int* __restrict__ src, const int* __restrict__ dst,
                           const float* __restrict__ exE, const float* __restrict__ den,
                           const float* __restrict__ hb, float* __restrict__ out, int E) {
    int i = blockIdx.x * blockDim.x + threadIdx.x;
    if (i >= E * 64) return;
    int e = i >> 6;
    int f4 = (i & 63) << 2;
    int h = f4 >> 6;
    int s = src[e], d = dst[e];
    float coef = exE[e * 4 + h] / den[d * 4 + h];
    float4 hv = *(const float4*)(hb + (size_t)s * FGn + f4);
    float* o = out + (size_t)d * FGn + f4;
    atomicAdd(o + 0, coef * hv.x);
    atomicAdd(o + 1, coef * hv.y);
    atomicAdd(o + 2, coef * hv.z);
    atomicAdd(o + 3, coef * hv.w);
}

// ---------------------------------------------------------------- host side
static inline int cdiv(int a, int b) { return (a + b - 1) / b; }

static void gemm_nn(const float* A, const float* W, const float* bias, float* C,
                    int n, int K, int F, int act, hipStream_t s) {
    int tiles = (n >> 4) * (F >> 4);
    gemm_wmma_nn<<<cdiv(tiles, TPB / 32), TPB, 0, s>>>(A, W, bias, C, n, K, F, act);
}

static void run_side(const int* src, const int* dst, const float* ew,
                     const float* x_in, const float* beta,
                     const float* gcnW, const float* gcnB,
                     const float* gatW, const float* attS, const float* attD, const float* gatB,
                     const float* l1W, const float* l1b,
                     const float* l2W, const float* l2b,
                     const float* l3W, const float* l3b,
                     int n, int E,
                     float* dinv, float* T, float* A, float* B, float* C,
                     float* as_, float* ad_, float* emax, float* den, float* exs, float* exE,
                     hipStream_t s) {
    const int nF = n * FGn;
    // degrees + symmetric norm
    k_fill<<<cdiv(n, TPB), TPB, 0, s>>>(dinv, 1.0f, n);          // self-loop weight 1
    k_deg<<<cdiv(E, TPB), TPB, 0, s>>>(dst, ew, dinv, E);
    k_dinv<<<cdiv(n, TPB), TPB, 0, s>>>(dinv, n);

    // GCN layer 1: x_in -> A
    gemm_nn(x_in, gcnW, nullptr, T, n, FGn, FGn, 0, s);
    k_gcn_self<<<cdiv(n * 64, TPB), TPB, 0, s>>>(T, dinv, A, n);
    k_gcn_scatter<<<cdiv(E * 64, TPB), TPB, 0, s>>>(src, dst, ew, dinv, T, A, E);
    k_bias_act<<<cdiv(nF, TPB), TPB, 0, s>>>(A, gcnB, nF, FGn);

    // GAT: A -> B
    gemm_nn(A, gatW, nullptr, T, n, FGn, FGn, 0, s);
    k_att<<<cdiv(n * 4, TPB), TPB, 0, s>>>(T, attS, attD, as_, ad_, n);
    k_emax_init<<<cdiv(n * 4, TPB), TPB, 0, s>>>(as_, ad_, emax, n);
    k_edge_max<<<cdiv(E, TPB), TPB, 0, s>>>(src, dst, as_, ad_, emax, E);
    k_self_exp<<<cdiv(n * 4, TPB), TPB, 0, s>>>(as_, ad_, emax, exs, den, n);
    k_edge_exp<<<cdiv(E, TPB), TPB, 0, s>>>(src, dst, as_, ad_, emax, exE, den, E);
    k_self_agg<<<cdiv(n * 64, TPB), TPB, 0, s>>>(T, exs, den, B, n);
    k_edge_agg<<<cdiv(E * 64, TPB), TPB, 0, s>>>(src, dst, exE, den, T, B, E);
    k_bias_act<<<cdiv(nF, TPB), TPB, 0, s>>>(B, gatB, nF, FGn);   // B = X1

    // GCN layer 2: B -> A (X2)
    gemm_nn(B, gcnW + FGn * FGn, nullptr, T, n, FGn, FGn, 0, s);
    k_gcn_self<<<cdiv(n * 64, TPB), TPB, 0, s>>>(T, dinv, A, n);
    k_gcn_scatter<<<cdiv(E * 64, TPB), TPB, 0, s>>>(src, dst, ew, dinv, T, A, E);
    k_bias_act<<<cdiv(nF, TPB), TPB, 0, s>>>(A, gcnB + FGn, nF, FGn);

    // GCN layer 3: A -> C (X3)
    gemm_nn(A, gcnW + 2 * FGn * FGn, nullptr, T, n, FGn, FGn, 0, s);
    k_gcn_self<<<cdiv(n * 64, TPB), TPB, 0, s>>>(T, dinv, C, n);
    k_gcn_scatter<<<cdiv(E * 64, TPB), TPB, 0, s>>>(src, dst, ew, dinv, T, C, E);
    k_bias_act<<<cdiv(nF, TPB), TPB, 0, s>>>(C, gcnB + 2 * FGn, nF, FGn);

    // combine: A = b0*X1 + b1*X2 + (1-b0-b1)*X3
    k_combine<<<cdiv(nF, TPB), TPB, 0, s>>>(A, B, C, beta, nF);

    // MLP: A -> T(256) -> B(128) -> C(64)
    gemm_nn(A, l1W, l1b, T, n, 256, 256, 1, s);
    gemm_nn(T, l2W, l2b, B, n, 256, 128, 1, s);
    gemm_nn(B, l3W, l3b, C, n, 128, 64, 1, s);
    // final per-side embedding lives in C [n,64]
}

extern "C" void kernel_launch(void* const* d_in, const int* in_sizes, int n_in,
                              void* d_out, int out_size, void* d_ws, size_t ws_size,
                              hipStream_t stream) {
    const int*   ei_m   = (const int*)d_in[0];
    const int*   ei_d   = (const int*)d_in[1];
    const float* ew_m   = (const float*)d_in[2];
    const float* ew_d   = (const float*)d_in[3];
    const float* x_m    = (const float*)d_in[4];
    const float* x_d    = (const float*)d_in[5];
    const float* beta1  = (const float*)d_in[6];
    const float* beta2  = (const float*)d_in[7];
    const float* gcn_Wx = (const float*)d_in[8];
    const float* gcn_bx = (const float*)d_in[9];
    const float* gcn_Wy = (const float*)d_in[10];
    const float* gcn_by = (const float*)d_in[11];
    const float* gat_Wx = (const float*)d_in[12];
    const float* att_xs = (const float*)d_in[13];
    const float* att_xd = (const float*)d_in[14];
    const float* gat_bx = (const float*)d_in[15];
    const float* gat_Wy = (const float*)d_in[16];
    const float* att_ys = (const float*)d_in[17];
    const float* att_yd = (const float*)d_in[18];
    const float* gat_by = (const float*)d_in[19];
    const float* lx1_W  = (const float*)d_in[20];
    const float* lx1_b  = (const float*)d_in[21];
    const float* lx2_W  = (const float*)d_in[22];
    const float* lx2_b  = (const float*)d_in[23];
    const float* lx3_W  = (const float*)d_in[24];
    const float* lx3_b  = (const float*)d_in[25];
    const float* ly1_W  = (const float*)d_in[26];
    const float* ly1_b  = (const float*)d_in[27];
    const float* ly2_W  = (const float*)d_in[28];
    const float* ly2_b  = (const float*)d_in[29];
    const float* ly3_W  = (const float*)d_in[30];
    const float* ly3_b  = (const float*)d_in[31];
    float* out = (float*)d_out;

    // bump allocator over d_ws (floats)
    float* w = (float*)d_ws;
    size_t off = 0;
    auto alloc = [&](size_t cnt) { float* p = w + off; off += cnt; return p; };

    // m side buffers
    float* dinv_m = alloc(Mn);
    float* T_m = alloc((size_t)Mn * FGn);
    float* A_m = alloc((size_t)Mn * FGn);
    float* B_m = alloc((size_t)Mn * FGn);
    float* C_m = alloc((size_t)Mn * FGn);
    float* as_m = alloc(Mn * 4); float* ad_m = alloc(Mn * 4);
    float* emax_m = alloc(Mn * 4); float* den_m = alloc(Mn * 4); float* exs_m = alloc(Mn * 4);
    float* exE_m = alloc((size_t)EMn * 4);
    // d side buffers
    float* dinv_d = alloc(Dn);
    float* T_d = alloc((size_t)Dn * FGn);
    float* A_d = alloc((size_t)Dn * FGn);
    float* B_d = alloc((size_t)Dn * FGn);
    float* C_d = alloc((size_t)Dn * FGn);
    float* as_d = alloc(Dn * 4); float* ad_d = alloc(Dn * 4);
    float* emax_d = alloc(Dn * 4); float* den_d = alloc(Dn * 4); float* exs_d = alloc(Dn * 4);
    float* exE_d = alloc((size_t)EDn * 4);
    (void)ws_size; (void)in_sizes; (void)n_in; (void)out_size;

    run_side(ei_m, ei_m + EMn, ew_m, x_m, beta1,
             gcn_Wx, gcn_bx, gat_Wx, att_xs, att_xd, gat_bx,
             lx1_W, lx1_b, lx2_W, lx2_b, lx3_W, lx3_b,
             Mn, EMn, dinv_m, T_m, A_m, B_m, C_m,
             as_m, ad_m, emax_m, den_m, exs_m, exE_m, stream);

    run_side(ei_d, ei_d + EDn, ew_d, x_d, beta2,
             gcn_Wy, gcn_by, gat_Wy, att_ys, att_yd, gat_by,
             ly1_W, ly1_b, ly2_W, ly2_b, ly3_W, ly3_b,
             Dn, EDn, dinv_d, T_d, A_d, B_d, C_d,
             as_d, ad_d, emax_d, den_d, exs_d, exE_d, stream);

    // final: out[M,D] = x[M,64] @ y[D,64]^T  (store-bound: 384 MB)
    int tiles = (Mn >> 4) * (Dn >> 4);
    gemm_wmma_nt<<<cdiv(tiles, TPB / 32), TPB, 0, stream>>>(C_m, C_d, out, Mn, Dn, 64);
}